// Gradient_Net_21036749816348
// MI455X (gfx1250) — compile-verified
//
#include <hip/hip_runtime.h>
#include <hip/hip_bf16.h>

// High-pass 3x3 stencil (k = 1/8 everywhere, -1 center), SAME zero padding,
// out = |conv(x,k)| + 1e-5, for x: (8,32,512,512) f32 -> 256 independent
// 512x512 images (channels folded into batch, like the reference).
//
// Memory-bound (~2 FLOP/byte; 537 MB total -> ~23us at 23.3 TB/s), so:
//  * Async global->LDS staging in 16B (b128) granules, aligned both sides.
//  * LDS halo tiling: 18 rows x 130 cols for a 128x16 output tile.
//  * Horizontal 3-sum S per-lane from LDS; vertical 3-sum on the matrix
//    pipe: Y = T*S with T = 16x16 tridiagonal ones via 4 chained
//    V_WMMA_F32_16X16X4_F32; tile-edge rows fixed up with 2 predicated adds.

typedef __attribute__((ext_vector_type(2))) float v2f;
typedef __attribute__((ext_vector_type(4))) float v4f;
typedef __attribute__((ext_vector_type(8))) float v8f;
typedef int v4i __attribute__((vector_size(4 * sizeof(int))));

typedef __attribute__((address_space(1))) int* as1_i32p;
typedef __attribute__((address_space(3))) int* as3_i32p;
typedef __attribute__((address_space(1))) v4i* as1_v4ip;
typedef __attribute__((address_space(3))) v4i* as3_v4ip;

#define IMG_H 512
#define IMG_W 512
#define TILE_W 128
#define TILE_H 16
#define HALO_H (TILE_H + 2)   // 18 rows
#define LDS_STRIDE 136        // floats; row stride 544 B (16B aligned)
#define COL0 4                // bulk cols live at LDS cols 4..131 (16B aligned)
#define LCOL 3                // left halo col  (global gc0-1)
#define RCOL (COL0 + TILE_W)  // right halo col (global gc0+128) = 132
#define QPR (TILE_W / 4)      // 32 quads per row
#define NTHREADS 256

#if __has_builtin(__builtin_amdgcn_global_load_async_to_lds_b32)
#define USE_ASYNC_LDS 1
#else
#define USE_ASYNC_LDS 0
#endif
#if __has_builtin(__builtin_amdgcn_global_load_async_to_lds_b128)
#define USE_ASYNC_B128 1
#else
#define USE_ASYNC_B128 0
#endif

#if __has_builtin(__builtin_amdgcn_wmma_f32_16x16x4_f32)
#define USE_WMMA 1
#else
#define USE_WMMA 0
#endif

// Stage one aligned 16-byte quad from global into LDS.
__device__ __forceinline__ void stage_quad(const float* g, float* l) {
#if USE_ASYNC_B128
  __builtin_amdgcn_global_load_async_to_lds_b128((as1_v4ip)g, (as3_v4ip)l, 0, 0);
#elif USE_ASYNC_LDS
#pragma unroll
  for (int k = 0; k < 4; ++k)
    __builtin_amdgcn_global_load_async_to_lds_b32((as1_i32p)(g + k),
                                                  (as3_i32p)(l + k), 0, 0);
#else
  *(v4f*)l = *(const v4f*)g;
#endif
}

__device__ __forceinline__ void stage_one(const float* g, float* l) {
#if USE_ASYNC_LDS
  __builtin_amdgcn_global_load_async_to_lds_b32((as1_i32p)g, (as3_i32p)l, 0, 0);
#else
  *l = *g;
#endif
}

__global__ __launch_bounds__(NTHREADS) void highpass3x3_wmma(
    const float* __restrict__ x, float* __restrict__ out) {
  __shared__ float tile[HALO_H * LDS_STRIDE];

  const int tid = threadIdx.x;
  const int z = blockIdx.z;                      // image index (n*c fold)
  const float* img = x + (size_t)z * IMG_H * IMG_W;
  float* oimg = out + (size_t)z * IMG_H * IMG_W;

  const int gr0 = blockIdx.y * TILE_H - 1;       // halo start row in image
  const int gcb = blockIdx.x * TILE_W;           // bulk start col in image

  // --- Stage 1: zero only the potentially-OOB strips --------------------
  // Top/bottom halo rows (full width) + left/right halo columns.
  if (tid < LDS_STRIDE) {
    tile[tid] = 0.0f;                                  // row 0
    tile[(HALO_H - 1) * LDS_STRIDE + tid] = 0.0f;      // row 17
  }
  if (tid < HALO_H) {
    tile[tid * LDS_STRIDE + LCOL] = 0.0f;              // left halo col
    tile[tid * LDS_STRIDE + RCOL] = 0.0f;              // right halo col
  }
  __syncthreads();

  // --- Stage 2: async-stage the tile ------------------------------------
  // Bulk: 18 rows x 32 quads of b128, fully aligned (global: gcb multiple of
  // 128; LDS: col 4 -> byte 16, row stride 544 B).
  for (int e = tid; e < HALO_H * QPR; e += NTHREADS) {   // 576 quads
    int r = e >> 5;                                      // e / QPR
    int q = e & (QPR - 1);
    int gr = gr0 + r;
    if (gr >= 0 && gr < IMG_H)
      stage_quad(img + (size_t)gr * IMG_W + gcb + 4 * q,
                 &tile[r * LDS_STRIDE + COL0 + 4 * q]);
  }
  // Side halo columns: 2 x 18 single floats.
  if (tid < 2 * HALO_H) {
    int side = (tid >= HALO_H) ? 1 : 0;
    int r = tid - side * HALO_H;
    int gr = gr0 + r;
    int gc = side ? (gcb + TILE_W) : (gcb - 1);
    if (gr >= 0 && gr < IMG_H && gc >= 0 && gc < IMG_W)
      stage_one(img + (size_t)gr * IMG_W + gc,
                &tile[r * LDS_STRIDE + (side ? RCOL : LCOL)]);
  }
#if USE_ASYNC_LDS
#if __has_builtin(__builtin_amdgcn_s_wait_asynccnt)
  __builtin_amdgcn_s_wait_asynccnt(0);
#else
  asm volatile("s_wait_asynccnt 0" ::: "memory");
#endif
#endif
  __syncthreads();

  // --- Stage 3: per-wave 16x16 sub-tile via WMMA -------------------------
  const int wave = tid >> 5;        // 0..7, one 16-wide column strip each
  const int lane = tid & 31;
  const int half = lane >> 4;       // lane-half selects K/M split per ISA layout
  const int n = lane & 15;          // output column within sub-tile
  const int sc = wave * 16;         // sub-tile column offset inside the 128 tile

  // Horizontal 3-sum at halo row hr, output column n of this sub-tile.
  // Output col n -> LDS center col COL0+sc+n; reads cols COL0+sc+n-1..+1.
  auto S = [&](int hr) -> float {
    int base = hr * LDS_STRIDE + sc + n + (COL0 - 1);
    return tile[base] + tile[base + 1] + tile[base + 2];
  };

  float sum9[8];

#if USE_WMMA
  // Y = T * S_interior, T tridiagonal ones, via 4 chained 16x16x4 f32 WMMAs.
  // A layout (16x4 f32): lane holds row M = lane%16; VGPR0 = cols {k, k+2},
  // VGPR1 = cols {k+1, k+3} split by lane-half. B (4x16) mirrors that on rows.
  v8f acc = {};
#pragma unroll
  for (int k = 0; k < 16; k += 4) {
    const int m = n;                  // A-matrix row held by this lane
    const int j0 = k + 2 * half;      // column/row index for .x
    const int j1 = k + 1 + 2 * half;  // column/row index for .y
    int d0 = m - j0;
    int d1 = m - j1;
    v2f a, b;
    a.x = (d0 >= -1 && d0 <= 1) ? 1.0f : 0.0f;
    a.y = (d1 >= -1 && d1 <= 1) ? 1.0f : 0.0f;
    b.x = S(j0 + 1);                  // interior row j0 -> halo row j0+1
    b.y = S(j1 + 1);
    acc = __builtin_amdgcn_wmma_f32_16x16x4_f32(
        /*neg_a=*/false, a, /*neg_b=*/false, b,
        /*c_mod=*/(short)0, acc, /*reuse_a=*/false, /*reuse_b=*/false);
  }
  // Halo fixups: row M=0 (acc[0], lane-half 0) needs S[-1] (halo row 0);
  // row M=15 (acc[7], lane-half 1) needs S[16] (halo row 17).
  const float s_top = S(0);
  const float s_bot = S(HALO_H - 1);
  acc[0] += half ? 0.0f : s_top;
  acc[7] += half ? s_bot : 0.0f;
#pragma unroll
  for (int v = 0; v < 8; ++v) sum9[v] = acc[v];
#else
  // VALU fallback: direct vertical 3-sum of horizontal sums.
#pragma unroll
  for (int v = 0; v < 8; ++v) {
    int M = v + 8 * half;             // matches C/D WMMA row layout
    sum9[v] = S(M) + S(M + 1) + S(M + 2);
  }
#endif

  // --- Stage 4: epilogue + coalesced stores ------------------------------
  // C/D layout: VGPR v, lanes 0-15 -> row M=v; lanes 16-31 -> row M=v+8.
  const int orow0 = blockIdx.y * TILE_H;
  const int ocol = gcb + sc + n;
#pragma unroll
  for (int v = 0; v < 8; ++v) {
    int M = v + 8 * half;
    float center = tile[(M + 1) * LDS_STRIDE + sc + n + COL0];
    float y = 0.125f * sum9[v] - 1.125f * center;
    oimg[(size_t)(orow0 + M) * IMG_W + ocol] = fabsf(y) + 1e-5f;
  }
}

extern "C" void kernel_launch(void* const* d_in, const int* in_sizes, int n_in,
                              void* d_out, int out_size, void* d_ws, size_t ws_size,
                              hipStream_t stream) {
  (void)n_in; (void)d_ws; (void)ws_size; (void)out_size;
  const float* x = (const float*)d_in[0];
  float* out = (float*)d_out;
  const int n_images = in_sizes[0] / (IMG_H * IMG_W);   // 8*32 = 256
  dim3 grid(IMG_W / TILE_W, IMG_H / TILE_H, n_images);  // (4, 32, 256)
  dim3 block(NTHREADS);
  hipLaunchKernelGGL(highpass3x3_wmma, grid, block, 0, stream, x, out);
}